// PFNetDense_75900662054929
// MI455X (gfx1250) — compile-verified
//
#include <hip/hip_runtime.h>
#include <hip/hip_bf16.h>

typedef float v2f __attribute__((ext_vector_type(2)));
typedef float v8f __attribute__((ext_vector_type(8)));

// Problem constants (match reference setup_inputs)
constexpr int B      = 8;
constexpr int N      = 25600;
constexpr int D_MSG  = 128;
constexpr int D_NODE = 256;
constexpr int BIN    = 128;        // BIN_SIZE
constexpr int NB     = N / BIN;    // 200 bins
constexpr int NPROJ  = NB / 2;     // 100 rotation columns
constexpr int KPAD   = 400;        // keys 0..398, padded
constexpr int CHUNK  = 256;
constexpr int NCHUNK = N / CHUNK;  // 100
constexpr float DIST_MULT = 0.1f;
constexpr float EPS = 1e-6f;
constexpr int XS = D_MSG + 4;      // padded LDS row stride (132): conflict-free b64 WMMA loads

// ---------------------------------------------------------------------------
// 1) LSH projection + argmax over [mul, -mul] -> integer bin key
// ---------------------------------------------------------------------------
__global__ __launch_bounds__(128)
void k_proj(const float* __restrict__ xmsg, const float* __restrict__ rot,
            const unsigned char* __restrict__ msk, int* __restrict__ key) {
  __shared__ float rotT[NPROJ * D_MSG];   // transposed: [j][d], contiguous d
  int tid = threadIdx.x;
  for (int i = tid; i < NPROJ * D_MSG; i += 128) {
    int d = i / NPROJ, j = i - d * NPROJ;
    rotT[j * D_MSG + d] = rot[i];
  }
  __syncthreads();

  int p = blockIdx.x * 128 + tid;         // global point index < B*N
  const float4* xr = (const float4*)(xmsg + (size_t)p * D_MSG);
  float4 x4[D_MSG / 4];
  #pragma unroll
  for (int i = 0; i < D_MSG / 4; ++i) x4[i] = xr[i];

  float bp = -INFINITY, bn = -INFINITY;   // best over mul / best over -mul
  int ip = 0, iq = 0;
  for (int j = 0; j < NPROJ; ++j) {
    const float4* rr = (const float4*)(rotT + j * D_MSG);  // broadcast reads
    float acc = 0.f;
    #pragma unroll
    for (int i = 0; i < D_MSG / 4; ++i) {
      float4 r = rr[i];
      acc += x4[i].x * r.x + x4[i].y * r.y + x4[i].z * r.z + x4[i].w * r.w;
    }
    if (acc > bp)  { bp = acc;  ip = j; }
    if (-acc > bn) { bn = -acc; iq = j; }
  }
  // cmul = [mul, -mul]; argmax takes first max -> positives win ties
  int idx = (bp >= bn) ? ip : (NPROJ + iq);
  key[p] = idx + (msk[p] ? 0 : (NB - 1));
}

// ---------------------------------------------------------------------------
// 2) per-(batch,chunk) key histogram
// ---------------------------------------------------------------------------
__global__ __launch_bounds__(CHUNK)
void k_hist(const int* __restrict__ key, int* __restrict__ hist) {
  __shared__ int cnt[KPAD];
  int tid = threadIdx.x;
  for (int i = tid; i < KPAD; i += CHUNK) cnt[i] = 0;
  __syncthreads();
  int g = blockIdx.x;                      // b*NCHUNK + c
  int b = g / NCHUNK, c = g - b * NCHUNK;
  int k = key[b * N + c * CHUNK + tid];
  atomicAdd(&cnt[k], 1);
  __syncthreads();
  for (int i = tid; i < KPAD; i += CHUNK)
    hist[(size_t)g * KPAD + i] = cnt[i];
}

// ---------------------------------------------------------------------------
// 3) per-batch: chunk-wise exclusive scan per key, then exclusive scan over
//    key totals; hist[b][c][k] becomes global start offset for that group.
// ---------------------------------------------------------------------------
__global__ __launch_bounds__(512)
void k_scan(int* __restrict__ hist) {
  __shared__ int totals[KPAD];
  __shared__ int start[KPAD];
  int b = blockIdx.x, k = threadIdx.x;
  if (k < KPAD) {
    int run = 0;
    for (int c = 0; c < NCHUNK; ++c) {
      size_t o = ((size_t)(b * NCHUNK + c)) * KPAD + k;
      int t = hist[o]; hist[o] = run; run += t;
    }
    totals[k] = run;
  }
  __syncthreads();
  if (k == 0) {
    int s = 0;
    for (int i = 0; i < KPAD; ++i) { start[i] = s; s += totals[i]; }
  }
  __syncthreads();
  if (k < KPAD) {
    int ks = start[k];
    for (int c = 0; c < NCHUNK; ++c)
      hist[((size_t)(b * NCHUNK + c)) * KPAD + k] += ks;
  }
}

// ---------------------------------------------------------------------------
// 4) stable scatter: pos = chunkStart + intra-chunk rank (stable == argsort)
//    also emits bins_split output (as float)
// ---------------------------------------------------------------------------
__global__ __launch_bounds__(CHUNK)
void k_scatter(const int* __restrict__ key, const int* __restrict__ hist,
               int* __restrict__ order, float* __restrict__ out_bins) {
  __shared__ int keys[CHUNK];
  int tid = threadIdx.x;
  int g = blockIdx.x;
  int b = g / NCHUNK, c = g - b * NCHUNK;
  int n = c * CHUNK + tid;
  int k = key[b * N + n];
  keys[tid] = k;
  __syncthreads();
  int rank = 0;
  for (int i = 0; i < tid; ++i) rank += (keys[i] == k);
  int pos = hist[(size_t)g * KPAD + k] + rank;
  order[b * N + pos] = n;
  out_bins[b * N + pos] = (float)n;
}

// ---------------------------------------------------------------------------
// 5) gather x_node rows (1 KB each) and msk_b; 4 rows per 256-thread block
// ---------------------------------------------------------------------------
__global__ __launch_bounds__(256)
void k_gather(const float* __restrict__ xnode, const unsigned char* __restrict__ msk,
              const int* __restrict__ order,
              float* __restrict__ out_node, float* __restrict__ out_msk) {
  int t = threadIdx.x;
  int r = blockIdx.x * 4 + (t >> 6);       // b*N + pos
  int c = t & 63;
  int b = r / N;
  int src = order[r];
  const float4* in4 = (const float4*)(xnode + ((size_t)b * N + src) * D_NODE);
  float4* o4 = (float4*)(out_node + (size_t)r * D_NODE);
  o4[c] = in4[c];
  if (c == 0) out_msk[r] = msk[b * N + src] ? 1.f : 0.f;
}

// ---------------------------------------------------------------------------
// 6) per-bin Gram (WMMA f32 16x16x4) + Gaussian distance kernel.
//    Rows gathered into LDS with GLOBAL_LOAD_ASYNC_TO_LDS_B128 (ASYNCcnt).
//    Note: masking X before the Gram is redundant — dm is multiplied by
//    m_i*m_j, so masked pairs are zeroed regardless of d; unmasked pairs see
//    identical x. Hence a pure async copy suffices.
// ---------------------------------------------------------------------------
__global__ __launch_bounds__(256)
void k_gram(const float* __restrict__ xmsg, const unsigned char* __restrict__ msk,
            const int* __restrict__ order, float* __restrict__ out_dm) {
  extern __shared__ float smem[];
  float* X  = smem;               // BIN x XS (padded)
  float* sq = smem + BIN * XS;    // BIN
  float* mv = sq + BIN;           // BIN

  int tid = threadIdx.x;
  int bin = blockIdx.x;           // b*NB + k
  int b = bin / NB;
  int kk = bin - b * NB;

  // async gather of raw rows into LDS (2 threads per row, 256 B each)
  {
    int p = tid >> 1, half = tid & 1;
    int src = order[(size_t)b * N + (size_t)kk * BIN + p];
    if (half == 0) mv[p] = msk[b * N + src] ? 1.f : 0.f;
    const float* grow = xmsg + ((size_t)b * N + src) * D_MSG + half * 64;
    unsigned lbase = (unsigned)(uintptr_t)(X + p * XS + half * 64);  // addr[31:0] = LDS offset
    #pragma unroll
    for (int i = 0; i < 16; ++i) {
      asm volatile("global_load_async_to_lds_b128 %0, %1, off"
                   :: "v"(lbase + (unsigned)(i * 16)), "v"(grow + i * 4)
                   : "memory");
    }
  }
  asm volatile("s_wait_asynccnt 0x0" ::: "memory");
  __syncthreads();

  // row squared norms (raw rows; see note above)
  if (tid < BIN) {
    const float* rw = X + tid * XS;
    float acc = 0.f;
    #pragma unroll 8
    for (int d = 0; d < D_MSG; ++d) acc += rw[d] * rw[d];
    sq[tid] = acc;
  }
  __syncthreads();

  // WMMA: wave w owns row strip i0 = 16*w; sweep 8 column tiles
  int lane = tid & 31;
  int i0   = (tid >> 5) * 16;
  int lrow = lane & 15;
  int cb   = (lane >> 4) << 1;    // 0 or 2 (K sub-columns per A/B layout)

  // hoist A fragments for all 32 k-steps (row strip fixed per wave)
  v2f afrag[32];
  #pragma unroll
  for (int ks = 0; ks < 32; ++ks)
    afrag[ks] = *(const v2f*)&X[(i0 + lrow) * XS + ks * 4 + cb];

  int ihi = i0 + ((lane >> 4) << 3);   // C/D layout: M = r (+8 for hi lanes)
  float* outb = out_dm + (size_t)bin * BIN * BIN;

  for (int jt = 0; jt < 8; ++jt) {
    int j0 = jt * 16;
    v8f acc = {0.f, 0.f, 0.f, 0.f, 0.f, 0.f, 0.f, 0.f};
    const float* brow = &X[(j0 + lrow) * XS + cb];
    #pragma unroll
    for (int ks = 0; ks < 32; ++ks) {
      v2f bf = *(const v2f*)(brow + ks * 4);
      acc = __builtin_amdgcn_wmma_f32_16x16x4_f32(
          false, afrag[ks], false, bf, (short)0, acc, false, false);
    }
    int jj = j0 + lrow;
    float sqj = sq[jj], mj = mv[jj];
    #pragma unroll
    for (int r = 0; r < 8; ++r) {
      int i = ihi + r;
      float g  = acc[r];
      float d2 = sq[i] - 2.f * g + sqj;
      float d  = sqrtf(fmaxf(d2, EPS));
      float v  = __expf(-DIST_MULT * d);
      v = fminf(fmaxf(v, 0.f), 1.f);
      outb[(size_t)i * BIN + jj] = v * mv[i] * mj;
    }
  }
}

// ---------------------------------------------------------------------------
extern "C" void kernel_launch(void* const* d_in, const int* in_sizes, int n_in,
                              void* d_out, int out_size, void* d_ws, size_t ws_size,
                              hipStream_t stream) {
  const float*         xmsg  = (const float*)d_in[0];
  const float*         xnode = (const float*)d_in[1];
  const unsigned char* msk   = (const unsigned char*)d_in[2];  // jax bool = 1 byte
  const float*         rot   = (const float*)d_in[3];
  float* out = (float*)d_out;

  // tuple outputs concatenated flat: bins_split, x_node_b, dm, msk_b
  size_t off_bins = 0;
  size_t off_node = (size_t)B * N;
  size_t off_dm   = off_node + (size_t)B * N * D_NODE;
  size_t off_msk  = off_dm + (size_t)B * NB * BIN * BIN;

  // workspace: keys | permutation | histogram
  int* key   = (int*)d_ws;
  int* order = key + (size_t)B * N;
  int* hist  = order + (size_t)B * N;      // B*NCHUNK*KPAD ints

  k_proj   <<<B * N / 128, 128, 0, stream>>>(xmsg, rot, msk, key);
  k_hist   <<<B * NCHUNK, CHUNK, 0, stream>>>(key, hist);
  k_scan   <<<B, 512, 0, stream>>>(hist);
  k_scatter<<<B * NCHUNK, CHUNK, 0, stream>>>(key, hist, order, out + off_bins);
  k_gather <<<B * N / 4, 256, 0, stream>>>(xnode, msk, order, out + off_node, out + off_msk);

  size_t lds = (size_t)(BIN * XS + 2 * BIN) * sizeof(float);   // ~68.6 KB < 320 KB/WGP
  k_gram   <<<B * NB, 256, lds, stream>>>(xmsg, msk, order, out + off_dm);
}